// GATModel_2379411882310
// MI455X (gfx1250) — compile-verified
//
#include <hip/hip_runtime.h>
#include <math.h>

typedef __attribute__((ext_vector_type(2))) float v2f;
typedef __attribute__((ext_vector_type(8))) float v8f;

#define HID 128   // H*C for both layers
#define NH  4     // heads
#define NC  32    // channels per head
#define NEG_SLOPE 0.2f

// ---------------------------------------------------------------- utilities

__device__ __forceinline__ void atomic_max_f32(float* addr, float val) {
    // Monotonic reinterpretation trick; correct for mixed signs given -inf init.
    if (val >= 0.0f) {
        atomicMax((int*)addr, __float_as_int(val));
    } else {
        atomicMin((unsigned int*)addr, (unsigned int)__float_as_int(val));
    }
}

__device__ __forceinline__ void get_edge(const int* __restrict__ ei, int e,
                                         int E, int& s, int& d) {
    if (e < E) { s = ei[e]; d = ei[E + e]; }
    else       { s = e - E; d = e - E; }       // appended self loops
}

__global__ void fill_f32(float* __restrict__ p, float v, int n) {
    int i = blockIdx.x * blockDim.x + threadIdx.x;
    if (i < n) p[i] = v;
}

// ---------------------------------------------------------------- GEMM (WMMA)
// C[n, Oc] = A[n, K] @ W[Oc, K]^T   (fp32, V_WMMA_F32_16X16X4_F32)
// Block = 128 threads (4 waves). Each wave computes a 16x32 strip (two 16x16
// column tiles, Oc=128 total), reusing the A fragment across both WMMAs.
// Rows are clamped (not predicated) so all loads are unconditional b64s;
// stores remain guarded, so clamped duplicate rows are never written.
template <int K>
__global__ void gemm_wmma_f32(const float* __restrict__ A,
                              const float* __restrict__ W,
                              float* __restrict__ C,
                              int n, int Oc) {
    const int wave = threadIdx.x >> 5;         // 0..3
    const int lane = threadIdx.x & 31;
    const int rowBase = blockIdx.x * 16;
    const int colBase = wave * 32;             // two 16-wide tiles per wave

    const int half = lane >> 4;                // 0: lanes 0-15, 1: lanes 16-31
    const int l15  = lane & 15;
    int row = rowBase + l15;
    if (row >= n) row = n - 1;                 // clamp; store-guarded below
    const int col0 = colBase + l15;
    const int col1 = col0 + 16;

    // fp32 A/B fragment layout: VGPR0 = K{0|2}, VGPR1 = K{1|3} per lane half.
    const float* Ap  = A + (size_t)row  * K + 2 * half;
    const float* Wp0 = W + (size_t)col0 * K + 2 * half;
    const float* Wp1 = W + (size_t)col1 * K + 2 * half;

    v8f acc0 = {}, acc1 = {};
#pragma unroll
    for (int k = 0; k < K; k += 4) {
        const v2f a  = *(const v2f*)(Ap  + k);
        const v2f b0 = *(const v2f*)(Wp0 + k);
        const v2f b1 = *(const v2f*)(Wp1 + k);
        acc0 = __builtin_amdgcn_wmma_f32_16x16x4_f32(
            false, a, false, b0, (short)0, acc0, false, false);
        acc1 = __builtin_amdgcn_wmma_f32_16x16x4_f32(
            false, a, false, b1, (short)0, acc1, false, false);
    }

    // C/D layout: VGPR r -> M = rowBase + r (+8 for lane half 1), N = col
    const int crow0 = rowBase + half * 8;
#pragma unroll
    for (int r = 0; r < 8; ++r) {
        const int crow = crow0 + r;
        if (crow < n) {
            C[(size_t)crow * Oc + col0] = acc0[r];
            C[(size_t)crow * Oc + col1] = acc1[r];
        }
    }
}

// ---------------------------------------------------------------- attention dots
// a_s[n,h] = sum_c xp[n,h,c]*att_src[h,c]; same for a_d.
__global__ void att_dot(const float* __restrict__ xp,
                        const float* __restrict__ att_s,
                        const float* __restrict__ att_d,
                        float* __restrict__ a_s, float* __restrict__ a_d,
                        int n) {
    int i = blockIdx.x * blockDim.x + threadIdx.x;     // over n*NH
    if (i >= n * NH) return;
    const int nn = i >> 2, h = i & 3;
    const float* v  = xp + (size_t)nn * HID + h * NC;
    const float* ws = att_s + h * NC;
    const float* wd = att_d + h * NC;
    float ss = 0.0f, sd = 0.0f;
#pragma unroll 8
    for (int c = 0; c < NC; ++c) { float x = v[c]; ss += x * ws[c]; sd += x * wd[c]; }
    a_s[i] = ss; a_d[i] = sd;
}

// ---------------------------------------------------------------- edge passes
__device__ __forceinline__ float edge_logit(const float* __restrict__ a_s,
                                            const float* __restrict__ a_d,
                                            int s, int d, int h) {
    float v = a_s[s * NH + h] + a_d[d * NH + h];
    return v > 0.0f ? v : NEG_SLOPE * v;               // leaky_relu
}

__global__ void edge_max(const int* __restrict__ ei,
                         const float* __restrict__ a_s,
                         const float* __restrict__ a_d,
                         float* __restrict__ m, int E, int ET) {
    int i = blockIdx.x * blockDim.x + threadIdx.x;     // over ET*NH
    if (i >= ET * NH) return;
    const int e = i >> 2, h = i & 3;
    int s, d; get_edge(ei, e, E, s, d);
    atomic_max_f32(&m[d * NH + h], edge_logit(a_s, a_d, s, d, h));
}

__global__ void edge_denom(const int* __restrict__ ei,
                           const float* __restrict__ a_s,
                           const float* __restrict__ a_d,
                           const float* __restrict__ m,
                           float* __restrict__ denom, int E, int ET) {
    int i = blockIdx.x * blockDim.x + threadIdx.x;
    if (i >= ET * NH) return;
    const int e = i >> 2, h = i & 3;
    int s, d; get_edge(ei, e, E, s, d);
    const float v = edge_logit(a_s, a_d, s, d, h);
    atomicAdd(&denom[d * NH + h], expf(v - m[d * NH + h]));
}

// Wave-per-edge aggregation: lanes <-> 32 channels, loop over heads.
// agg[dst,h,c] += xp[src,h,c] * alpha(e,h)
__global__ void edge_agg(const int* __restrict__ ei,
                         const float* __restrict__ xp,
                         const float* __restrict__ a_s,
                         const float* __restrict__ a_d,
                         const float* __restrict__ m,
                         const float* __restrict__ denom,
                         float* __restrict__ agg, int E, int ET) {
    const int wv   = (blockIdx.x * blockDim.x + threadIdx.x) >> 5;
    const int lane = threadIdx.x & 31;
    if (wv >= ET) return;
    int s, d; get_edge(ei, wv, E, s, d);
    const float* xs = xp  + (size_t)s * HID;
    float*       ag = agg + (size_t)d * HID;
#pragma unroll
    for (int h = 0; h < NH; ++h) {
        const float v = edge_logit(a_s, a_d, s, d, h);
        const float alpha = expf(v - m[d * NH + h]) / (denom[d * NH + h] + 1e-16f);
        atomicAdd(&ag[h * NC + lane], xs[h * NC + lane] * alpha);
    }
}

// ---------------------------------------------------------------- epilogues
__global__ void elu_bias_inplace(float* __restrict__ a,
                                 const float* __restrict__ bias, int total) {
    int i = blockIdx.x * blockDim.x + threadIdx.x;
    if (i >= total) return;
    float v = a[i] + bias[i & (HID - 1)];
    a[i] = v > 0.0f ? v : expm1f(v);                   // ELU(alpha=1)
}

__global__ void mean_heads_bias(const float* __restrict__ agg,
                                const float* __restrict__ bias,
                                float* __restrict__ out, int n) {
    int i = blockIdx.x * blockDim.x + threadIdx.x;     // over n*NC
    if (i >= n * NC) return;
    const int nn = i >> 5, c = i & 31;
    const float* a = agg + (size_t)nn * HID;
    out[i] = 0.25f * (a[c] + a[NC + c] + a[2 * NC + c] + a[3 * NC + c]) + bias[c];
}

// ---------------------------------------------------------------- launcher

static inline int cdiv(int a, int b) { return (a + b - 1) / b; }

extern "C" void kernel_launch(void* const* d_in, const int* in_sizes, int n_in,
                              void* d_out, int out_size, void* d_ws, size_t ws_size,
                              hipStream_t stream) {
    const float* x        = (const float*)d_in[0];
    const int*   ei       = (const int*)  d_in[1];
    const float* W1       = (const float*)d_in[2];
    const float* att_src1 = (const float*)d_in[3];
    const float* att_dst1 = (const float*)d_in[4];
    const float* bias1    = (const float*)d_in[5];
    const float* W2       = (const float*)d_in[6];
    const float* att_src2 = (const float*)d_in[7];
    const float* att_dst2 = (const float*)d_in[8];
    const float* bias2    = (const float*)d_in[9];
    float* out = (float*)d_out;

    const int IN = 64;
    const int N  = in_sizes[0] / IN;
    const int E  = in_sizes[1] / 2;
    const int ET = E + N;                      // with self loops

    // workspace partition (floats)
    float* xp1   = (float*)d_ws;               // N*HID
    float* agg1  = xp1  + (size_t)N * HID;     // N*HID (becomes h after ELU)
    float* xp2   = agg1 + (size_t)N * HID;     // N*HID
    float* agg2  = xp2  + (size_t)N * HID;     // N*HID
    float* a_s   = agg2 + (size_t)N * HID;     // N*NH
    float* a_d   = a_s  + (size_t)N * NH;      // N*NH
    float* mbuf  = a_d  + (size_t)N * NH;      // N*NH
    float* dnm   = mbuf + (size_t)N * NH;      // N*NH

    const int BT = 256;
    const int gRowTiles = cdiv(N, 16);
    const int gNH  = cdiv(N * NH, BT);
    const int gEH  = cdiv(ET * NH, BT);
    const int gEW  = cdiv(ET * 32, BT);
    const int gNF  = cdiv(N * HID, BT);
    const int gOut = cdiv(N * NC, BT);

    // ---------------- layer 1 ----------------
    gemm_wmma_f32<64><<<gRowTiles, 128, 0, stream>>>(x, W1, xp1, N, HID);
    att_dot<<<gNH, BT, 0, stream>>>(xp1, att_src1, att_dst1, a_s, a_d, N);
    fill_f32<<<gNH, BT, 0, stream>>>(mbuf, -INFINITY, N * NH);
    fill_f32<<<gNH, BT, 0, stream>>>(dnm, 0.0f, N * NH);
    fill_f32<<<gNF, BT, 0, stream>>>(agg1, 0.0f, N * HID);
    edge_max  <<<gEH, BT, 0, stream>>>(ei, a_s, a_d, mbuf, E, ET);
    edge_denom<<<gEH, BT, 0, stream>>>(ei, a_s, a_d, mbuf, dnm, E, ET);
    edge_agg  <<<gEW, BT, 0, stream>>>(ei, xp1, a_s, a_d, mbuf, dnm, agg1, E, ET);
    elu_bias_inplace<<<gNF, BT, 0, stream>>>(agg1, bias1, N * HID);   // agg1 -> h

    // ---------------- layer 2 ----------------
    gemm_wmma_f32<128><<<gRowTiles, 128, 0, stream>>>(agg1, W2, xp2, N, HID);
    att_dot<<<gNH, BT, 0, stream>>>(xp2, att_src2, att_dst2, a_s, a_d, N);
    fill_f32<<<gNH, BT, 0, stream>>>(mbuf, -INFINITY, N * NH);
    fill_f32<<<gNH, BT, 0, stream>>>(dnm, 0.0f, N * NH);
    fill_f32<<<gNF, BT, 0, stream>>>(agg2, 0.0f, N * HID);
    edge_max  <<<gEH, BT, 0, stream>>>(ei, a_s, a_d, mbuf, E, ET);
    edge_denom<<<gEH, BT, 0, stream>>>(ei, a_s, a_d, mbuf, dnm, E, ET);
    edge_agg  <<<gEW, BT, 0, stream>>>(ei, xp2, a_s, a_d, mbuf, dnm, agg2, E, ET);
    mean_heads_bias<<<gOut, BT, 0, stream>>>(agg2, bias2, out, N);
}